// CholeskyLKJImpl_76553497084289
// MI455X (gfx1250) — compile-verified
//
#include <hip/hip_runtime.h>
#include <hip/hip_bf16.h>

// CholeskyLKJ transform: B=16384 batches, N=64, TRIL=2016.
// Roofline: 132MB in + 268MB out => ~17us @ 23.3 TB/s; compute trimmed to
// ~3 fast transcendentals/element so VALU stays under the HBM bound.
// One wave32 per batch; lane t owns rows t and 63-t (exactly 63 elements/lane),
// processed in ONE uniform 63-iteration loop (100% lane utilization).
// CDNA5 paths: GLOBAL_LOAD_ASYNC_TO_LDS_B128 / GLOBAL_STORE_ASYNC_FROM_LDS_B128
// (ASYNCcnt DMA engine) with non-temporal hints (streaming data, touched once),
// fenced by s_wait_asynccnt / s_wait_dscnt.

#define B_TOTAL 16384
#define NMAT    64
#define TRILN   2016
#define BPB     2                 // batches (waves) per block
#define THREADS (BPB * 32)
#define GRID    (B_TOTAL / BPB)   // 8192 blocks

__device__ __forceinline__ void async_load_f4(unsigned lds_off, const void* gaddr) {
    // vdst = VGPR with LDS byte address, vaddr = 64-bit global address (GV mode)
    asm volatile("global_load_async_to_lds_b128 %0, %1, off th:TH_LOAD_NT"
                 :: "v"(lds_off), "v"(gaddr) : "memory");
}

__device__ __forceinline__ void async_store_f4(void* gaddr, unsigned lds_off) {
    asm volatile("global_store_async_from_lds_b128 %0, %1, off th:TH_STORE_NT"
                 :: "v"(gaddr), "v"(lds_off) : "memory");
}

__device__ __forceinline__ unsigned lds_addr_of(const void* p) {
    // generic pointer into LDS aperture: low 32 bits = wave-relative LDS address
    return (unsigned)(size_t)p;
}

__global__ __launch_bounds__(THREADS)
void lkj_main_kernel(const float* __restrict__ tril,
                     float* __restrict__ z,
                     float* __restrict__ partials) {
    __shared__ __align__(16) float s_in[BPB][2048];        // 2016 used
    __shared__ __align__(16) float s_out[BPB][NMAT * NMAT];
    __shared__ float s_red[BPB];

    const int wave  = threadIdx.x >> 5;
    const int lane  = threadIdx.x & 31;
    const int batch = blockIdx.x * BPB + wave;

    const float* gin  = tril + (size_t)batch * TRILN;
    float*       gout = z    + (size_t)batch * (NMAT * NMAT);

    // ---- stage packed tril (8064B) into LDS via async DMA ----
    const unsigned lds_in_base = lds_addr_of(&s_in[wave][0]);
    const int NF4 = TRILN / 4;  // 504 float4 chunks
    for (int i = lane; i < NF4; i += 32) {
        async_load_f4(lds_in_base + (unsigned)i * 16u, (const void*)(gin + i * 4));
    }

    // ---- zero the 16KB z tile with uniform vector LDS stores (overlaps DMA) ----
    float*  sout_w = &s_out[wave][0];
    float4* so4    = (float4*)sout_w;
    const float4 z4 = make_float4(0.f, 0.f, 0.f, 0.f);
    #pragma unroll 4
    for (int i = lane; i < (NMAT * NMAT) / 4; i += 32) so4[i] = z4;

    asm volatile("s_wait_asynccnt 0x0" ::: "memory");

    const float* sin_w = &s_in[wave][0];

    // lane owns row r1 = lane (r1 elems) then row r2 = 63-lane (r2 elems);
    // r1 + r2 = 63 -> one uniform loop, every lane busy every iteration.
    const int r1 = lane;
    const int r2 = NMAT - 1 - lane;
    const int base1 = r1 * (r1 - 1) / 2;   // packed rows are contiguous
    const int base2 = r2 * (r2 - 1) / 2;

    float c = 0.0f;    // constraint partial
    float s = 1.0f;    // running prod_{k<j} (1 - tanh^2)

    for (int k = 0; k < NMAT - 1; ++k) {
        if (k == r1) {                       // first row done: diag, reset scan
            sout_w[r1 * NMAT + r1] = sqrtf(s);
            s = 1.0f;
        }
        const bool first = (k < r1);
        const int  row   = first ? r1 : r2;
        const int  jj    = first ? k : (k - r1);
        const int  ib    = first ? base1 : base2;

        const float x = sin_w[ib + jj];
        const float a = fabsf(x);
        const float e = __expf(-2.0f * a);                 // shared by tanh & logcosh
        float t = __fdividef(1.0f - e, 1.0f + e);          // tanh(|x|)
        t = copysignf(t, x);
        // log cosh(x) = |x| + log1p(e) - log 2 ; 1+e in [1,2] -> fast log is safe
        const float lc = a + __logf(1.0f + e) - 0.69314718056f;
        c = fmaf(-2.0f, lc, c);
        c = fmaf(0.5f, __logf(s), c);                      // 0.5 * sum log s[i,j]
        sout_w[row * NMAT + jj] = t * sqrtf(s);            // strict lower entry
        s *= (1.0f - t * t);
    }
    sout_w[r2 * NMAT + r2] = sqrtf(s);                     // diag of second row

    // ---- deterministic constraint reduction ----
    #pragma unroll
    for (int m = 16; m >= 1; m >>= 1) c += __shfl_xor(c, m, 32);
    if (lane == 0) s_red[wave] = c;
    __syncthreads();
    if (threadIdx.x == 0) {
        float acc = 0.0f;
        #pragma unroll
        for (int w = 0; w < BPB; ++w) acc += s_red[w];
        partials[blockIdx.x] = acc;
    }

    // ---- DMA the 16KB z tile out of LDS, coalesced, non-temporal ----
    asm volatile("s_wait_dscnt 0x0" ::: "memory");   // LDS writes must land first
    const unsigned lds_out_base = lds_addr_of(sout_w);
    for (int i = lane; i < (NMAT * NMAT) / 4; i += 32) {   // exactly 32 per lane
        async_store_f4((void*)(gout + i * 4), lds_out_base + (unsigned)i * 16u);
    }
    asm volatile("s_wait_asynccnt 0x0" ::: "memory");
}

__global__ __launch_bounds__(256)
void lkj_reduce_kernel(const float* __restrict__ partials, int n,
                       float* __restrict__ out) {
    __shared__ float red[8];
    float s = 0.0f;
    for (int i = threadIdx.x; i < n; i += 256) s += partials[i];  // fixed order
    #pragma unroll
    for (int m = 16; m >= 1; m >>= 1) s += __shfl_xor(s, m, 32);
    const int wave = threadIdx.x >> 5, lane = threadIdx.x & 31;
    if (lane == 0) red[wave] = s;
    __syncthreads();
    if (threadIdx.x == 0) {
        float acc = 0.0f;
        #pragma unroll
        for (int w = 0; w < 8; ++w) acc += red[w];
        *out = acc;
    }
}

extern "C" void kernel_launch(void* const* d_in, const int* in_sizes, int n_in,
                              void* d_out, int out_size, void* d_ws, size_t ws_size,
                              hipStream_t stream) {
    const float* tril = (const float*)d_in[0];
    float* zout  = (float*)d_out;                        // B*64*64 floats
    float* cons  = zout + (size_t)B_TOTAL * NMAT * NMAT; // scalar after z
    float* parts = (float*)d_ws;                         // GRID floats (32KB)

    lkj_main_kernel<<<GRID, THREADS, 0, stream>>>(tril, zout, parts);
    lkj_reduce_kernel<<<1, 256, 0, stream>>>(parts, GRID, cons);
}